// AttNet_1065151889466
// MI455X (gfx1250) — compile-verified
//
#include <hip/hip_runtime.h>
#include <hip/hip_bf16.h>

#define USE_ASYNC_FILL 1   // gfx1250 GLOBAL_LOAD_ASYNC_TO_LDS fill path

typedef float v2f __attribute__((ext_vector_type(2)));
typedef float v8f __attribute__((ext_vector_type(8)));

// ---------------- problem constants ----------------
#define BSZ   2
#define TT    3
#define CPT   7
#define NPT   130000
#define FP    64
#define BEVH  512
#define BEVW  512
#define RVH   64
#define RVW   2048
#define BEVHW (BEVH*BEVW)
#define RVHW  (RVH*RVW)

// ---------------- workspace layout (float offsets) ----------------
static const size_t H_OFF    = 0;
static const size_t H_SZ     = (size_t)BSZ*TT*FP*NPT;
static const size_t BEV_OFF  = H_OFF + H_SZ;
static const size_t BEV_SZ   = (size_t)BSZ*TT*FP*BEVHW;
static const size_t RV_OFF   = BEV_OFF + BEV_SZ;
static const size_t RV_SZ    = (size_t)BSZ*FP*RVHW;
static const size_t BEVF_OFF = RV_OFF + RV_SZ;
static const size_t BEVF_SZ  = (size_t)BSZ*FP*BEVHW;
static const size_t RVF_OFF  = BEVF_OFF + BEVF_SZ;
static const size_t RVF_SZ   = (size_t)BSZ*FP*RVHW;
static const size_t PAB_OFF  = RVF_OFF + RVF_SZ;      // packed A, BEV: 64*1728
static const size_t PAB_SZ   = (size_t)64*192*9;
static const size_t PAR_OFF  = PAB_OFF + PAB_SZ;      // packed A, RV: 64*576
static const size_t PAR_SZ   = (size_t)64*64*9;

// ---------------- zero the scatter grids ----------------
__global__ void k_zero(float4* __restrict__ p, size_t n4) {
  size_t i = (size_t)blockIdx.x * blockDim.x + threadIdx.x;
  size_t stride = (size_t)gridDim.x * blockDim.x;
  float4 z; z.x = 0.f; z.y = 0.f; z.z = 0.f; z.w = 0.f;
  for (; i < n4; i += stride) p[i] = z;
}

// ---------------- per-point MLP: 7 -> 64 -> 64, ReLU ----------------
__global__ void k_mlp(const float* __restrict__ x, const float* __restrict__ w1,
                      const float* __restrict__ w2, float* __restrict__ h) {
  int i = blockIdx.x * blockDim.x + threadIdx.x;
  if (i >= BSZ * TT * NPT) return;
  int bt = i / NPT, n = i % NPT;
  float xin[CPT];
#pragma unroll
  for (int c = 0; c < CPT; ++c) xin[c] = x[((size_t)bt * CPT + c) * NPT + n];
  float a1[FP];
#pragma unroll
  for (int o = 0; o < FP; ++o) {
    float s = 0.f;
#pragma unroll
    for (int c = 0; c < CPT; ++c) s = fmaf(w1[o * CPT + c], xin[c], s);
    a1[o] = fmaxf(s, 0.f);
  }
  float* hp = h + (size_t)bt * FP * NPT + n;
  for (int o = 0; o < FP; ++o) {
    float s = 0.f;
#pragma unroll
    for (int c = 0; c < FP; ++c) s = fmaf(w2[o * FP + c], a1[c], s);
    hp[(size_t)o * NPT] = fmaxf(s, 0.f);
  }
}

// ---------------- scatter-max (uint bit trick, values >= 0) ----------------
__global__ void k_scatter_bev(const float* __restrict__ h, const int* __restrict__ coord,
                              unsigned int* __restrict__ bev) {
  int i = blockIdx.x * blockDim.x + threadIdx.x;
  if (i >= BSZ * TT * NPT) return;
  int bt = i / NPT, n = i % NPT;
  const int* pc = coord + ((size_t)bt * NPT + n) * 3;
  int idx = pc[0] * BEVW + pc[1];
  const float* hp = h + (size_t)bt * FP * NPT + n;
  unsigned int* gp = bev + (size_t)bt * FP * BEVHW + idx;
#pragma unroll 8
  for (int c = 0; c < FP; ++c)
    atomicMax(gp + (size_t)c * BEVHW, __float_as_uint(hp[(size_t)c * NPT]));
}

__global__ void k_scatter_rv(const float* __restrict__ h, const int* __restrict__ sph,
                             unsigned int* __restrict__ rv) {
  int i = blockIdx.x * blockDim.x + threadIdx.x;
  if (i >= BSZ * NPT) return;
  int b = i / NPT, n = i % NPT;
  const int* ps = sph + ((size_t)(b * TT) * NPT + n) * 2;   // t = 0
  int idx = ps[0] * RVW + ps[1];
  const float* hp = h + (size_t)(b * TT) * FP * NPT + n;    // h_cur
  unsigned int* gp = rv + (size_t)b * FP * RVHW + idx;
#pragma unroll 8
  for (int c = 0; c < FP; ++c)
    atomicMax(gp + (size_t)c * RVHW, __float_as_uint(hp[(size_t)c * NPT]));
}

// ---------------- bev_t -= base for t = 1,2 (residual planes in place) ----------------
__global__ void k_resid(float* __restrict__ bev) {
  const size_t per_t = (size_t)FP * BEVHW;
  const size_t per_b = 2 * per_t;
  const size_t total = (size_t)BSZ * per_b;
  size_t stride = (size_t)gridDim.x * blockDim.x;
  for (size_t e = (size_t)blockIdx.x * blockDim.x + threadIdx.x; e < total; e += stride) {
    size_t b = e / per_b, r = e % per_b;
    size_t t = r / per_t + 1, cp = r % per_t;
    bev[((size_t)(b * TT + t)) * per_t + cp] -= bev[((size_t)(b * TT)) * per_t + cp];
  }
}

// ---------------- pack conv weights into WMMA A-fragment order ----------------
// K enumeration: slot s = ch*576 + r*64 + cl  (ch = cin/64 chunk, r = 3x3 tap, cl = cin%64)
template <int CIN>
__global__ void k_packA(const float* __restrict__ wt, float* __restrict__ pa) {
  const int K9 = CIN * 9;
  int i = blockIdx.x * blockDim.x + threadIdx.x;     // pair index
  if (i >= 32 * K9) return;                          // 4 mt * (K9/4) * 32 lanes
  int lane = i & 31;
  int t = i >> 5;
  int kq = t % (K9 / 4);
  int mt = t / (K9 / 4);
  int half = lane >> 4, l16 = lane & 15;
  int m = mt * 16 + l16;
  int s0 = kq * 4 + 2 * half;
#pragma unroll
  for (int e = 0; e < 2; ++e) {
    int s = s0 + e;
    int ch = s / 576;
    int rem = s % 576;
    int r = rem >> 6;
    int cl = rem & 63;
    int cin = ch * 64 + cl;
    pa[(size_t)i * 2 + e] = wt[(size_t)m * K9 + cin * 9 + r];
  }
}

// ---------------- implicit-GEMM 3x3 conv via V_WMMA_F32_16X16X4_F32 ----------------
// Block: 256 thr = 8 waves = 4 mtiles x 2 tile-groups; block owns 64 pixels of one row.
// Each wave computes 2 pixel tiles (2 accumulators) sharing one packed-A b64 load/step.
// Input window (3 rows x 66 cols x 64-ch chunk) staged in LDS with channel pairs
// interleaved so each WMMA B fragment is one ds_load_b64. In-bounds elements are DMA'd
// memory->LDS with GLOBAL_LOAD_ASYNC_TO_LDS_B32 (ASYNCcnt); OOB halo zeroed via ds_store.
template <int CIN, int H, int W, bool ISBEV>
__global__ void __launch_bounds__(256) k_conv_wmma(const float* __restrict__ in,
                                                   const float* __restrict__ pa,
                                                   float* __restrict__ out) {
  const int K9 = CIN * 9;
  const int NCH = CIN / 64;                 // 64-channel chunks
  const int XB = W / 64;
  __shared__ float lds[32 * 3 * 68 * 2];    // [cpair][dy][xcol pad 68][2] = 52224 B

  int xb = blockIdx.x % XB;
  int y  = (blockIdx.x / XB) % H;
  int b  = blockIdx.x / (XB * H);
  int x0 = xb * 64;

  int lane = threadIdx.x & 31;
  int wib  = threadIdx.x >> 5;
  int mt = wib & 3;
  int tg = wib >> 2;                        // tile group: pixel tiles 2*tg, 2*tg+1
  int half = lane >> 4;
  int l16  = lane & 15;
  int xcolbase = tg * 32 + l16 + 1;         // +1 for left halo column

  const v2f* pA = (const v2f*)pa + (size_t)mt * (K9 / 4) * 32 + lane;
  v8f acc0 = {0.f, 0.f, 0.f, 0.f, 0.f, 0.f, 0.f, 0.f};
  v8f acc1 = {0.f, 0.f, 0.f, 0.f, 0.f, 0.f, 0.f, 0.f};

  for (int ch = 0; ch < NCH; ++ch) {
    __syncthreads();
    // ---- fill: 64 channels x 3 rows x 66 cols ----
    const float* cb;
    if constexpr (ISBEV)
      cb = in + ((size_t)(b * TT + ch) * FP) * (size_t)(H * W);
    else
      cb = in + ((size_t)b * FP) * (size_t)(H * W);
    for (int e = threadIdx.x; e < 64 * 3 * 66; e += 256) {
      int cl   = e / 198;
      int rem  = e - cl * 198;
      int dy1  = rem / 66;
      int xcol = rem - dy1 * 66;
      int yy = y + dy1 - 1;
      int xx = x0 + xcol - 1;
      float* lp = &lds[(((cl >> 1) * 3 + dy1) * 68 + xcol) * 2 + (cl & 1)];
      bool inb = ((unsigned)yy < (unsigned)H) && ((unsigned)xx < (unsigned)W);
#if USE_ASYNC_FILL
      if (inb) {
        const float* gp = &cb[(size_t)cl * (H * W) + (size_t)yy * W + xx];
        unsigned int ldsaddr = (unsigned int)(unsigned long long)lp;  // flat low 32b = LDS addr
        unsigned long long gaddr = (unsigned long long)gp;
        asm volatile("global_load_async_to_lds_b32 %0, %1, off"
                     :: "v"(ldsaddr), "v"(gaddr) : "memory");
      } else {
        *lp = 0.f;
      }
#else
      float v = 0.f;
      if (inb) v = cb[(size_t)cl * (H * W) + (size_t)yy * W + xx];
      *lp = v;
#endif
    }
#if USE_ASYNC_FILL
    asm volatile("s_wait_asynccnt 0x0" ::: "memory");
#endif
    __syncthreads();
    // ---- WMMA over this chunk: K = 9 taps x 64 channels ----
    int kq = ch * 144;
#pragma unroll
    for (int r = 0; r < 9; ++r) {
      int dy1 = r / 3, dx = r % 3 - 1;
      int xc = xcolbase + dx;               // tile0 column; tile1 = +16
#pragma unroll 4
      for (int c0 = 0; c0 < 64; c0 += 4) {
        int cpair = (c0 >> 1) + half;       // channels c0+2*half, +1
        v2f av = pA[(size_t)kq * 32];
        const float* lb = &lds[((cpair * 3 + dy1) * 68) * 2];
        v2f bv0 = *(const v2f*)(lb + (size_t)xc * 2);
        v2f bv1 = *(const v2f*)(lb + (size_t)(xc + 16) * 2);
        acc0 = __builtin_amdgcn_wmma_f32_16x16x4_f32(
            false, av, false, bv0, (short)0, acc0, false, false);
        acc1 = __builtin_amdgcn_wmma_f32_16x16x4_f32(
            false, av, false, bv1, (short)0, acc1, false, false);
        ++kq;
      }
    }
  }

  // ---- ReLU + store: C/D layout VGPR j -> M = j + 8*half, N = lane%16 ----
  size_t orow = (size_t)y * W + (size_t)x0 + tg * 32 + l16;
#pragma unroll
  for (int j = 0; j < 8; ++j) {
    int m = mt * 16 + j + 8 * half;
    size_t base = ((size_t)b * FP + m) * (size_t)(H * W) + orow;
    out[base]      = fmaxf(acc0[j], 0.f);
    out[base + 16] = fmaxf(acc1[j], 0.f);
  }
}

// ---------------- gather + fuse MLP 192 -> 64 -> 3 ----------------
__global__ void k_fuse(const float* __restrict__ h, const float* __restrict__ bevf,
                       const float* __restrict__ rvf, const int* __restrict__ coord,
                       const int* __restrict__ sph, const float* __restrict__ wf,
                       const float* __restrict__ wp, float* __restrict__ out) {
  int i = blockIdx.x * blockDim.x + threadIdx.x;
  if (i >= BSZ * NPT) return;
  int b = i / NPT, n = i % NPT;
  const int* pc = coord + ((size_t)(b * TT) * NPT + n) * 3;
  int bidx = pc[0] * BEVW + pc[1];
  const int* ps = sph + ((size_t)(b * TT) * NPT + n) * 2;
  int ridx = ps[0] * RVW + ps[1];

  const float* hp = h + (size_t)(b * TT) * FP * NPT + n;
  const float* bp = bevf + (size_t)b * FP * BEVHW + bidx;
  const float* rp = rvf + (size_t)b * FP * RVHW + ridx;
  float f0[FP], f1[FP], f2[FP];
#pragma unroll
  for (int c = 0; c < FP; ++c) {
    f0[c] = hp[(size_t)c * NPT];
    f1[c] = bp[(size_t)c * BEVHW];
    f2[c] = rp[(size_t)c * RVHW];
  }
  float p0 = 0.f, p1 = 0.f, p2 = 0.f;
  for (int o = 0; o < FP; ++o) {
    const float* w = wf + o * 192;
    float s = 0.f;
#pragma unroll
    for (int c = 0; c < FP; ++c) s = fmaf(w[c], f0[c], s);
#pragma unroll
    for (int c = 0; c < FP; ++c) s = fmaf(w[64 + c], f1[c], s);
#pragma unroll
    for (int c = 0; c < FP; ++c) s = fmaf(w[128 + c], f2[c], s);
    s = fmaxf(s, 0.f);
    p0 = fmaf(wp[o], s, p0);
    p1 = fmaf(wp[64 + o], s, p1);
    p2 = fmaf(wp[128 + o], s, p2);
  }
  out[((size_t)b * 3 + 0) * NPT + n] = p0;
  out[((size_t)b * 3 + 1) * NPT + n] = p1;
  out[((size_t)b * 3 + 2) * NPT + n] = p2;
}

// ---------------- launch ----------------
extern "C" void kernel_launch(void* const* d_in, const int* in_sizes, int n_in,
                              void* d_out, int out_size, void* d_ws, size_t ws_size,
                              hipStream_t stream) {
  (void)in_sizes; (void)n_in; (void)out_size; (void)ws_size;
  const float* x     = (const float*)d_in[0];
  const int*   coord = (const int*)d_in[1];
  const int*   sph   = (const int*)d_in[2];
  const float* w1    = (const float*)d_in[3];
  const float* w2    = (const float*)d_in[4];
  const float* kb    = (const float*)d_in[5];
  const float* kr    = (const float*)d_in[6];
  const float* wf    = (const float*)d_in[7];
  const float* wp    = (const float*)d_in[8];
  float* out = (float*)d_out;
  float* ws  = (float*)d_ws;

  float* h    = ws + H_OFF;
  float* bev  = ws + BEV_OFF;   // rv grid contiguous right after bev
  float* rv   = ws + RV_OFF;
  float* bevf = ws + BEVF_OFF;
  float* rvf  = ws + RVF_OFF;
  float* pab  = ws + PAB_OFF;
  float* par  = ws + PAR_OFF;

  // 1) zero both scatter grids (contiguous region)
  size_t zero4 = (BEV_SZ + RV_SZ) / 4;
  k_zero<<<8192, 256, 0, stream>>>((float4*)bev, zero4);
  // 1b) pack conv weights into WMMA A-fragment order
  k_packA<192><<<(32 * 192 * 9 + 255) / 256, 256, 0, stream>>>(kb, pab);
  k_packA<64><<<(32 * 64 * 9 + 255) / 256, 256, 0, stream>>>(kr, par);
  // 2) point MLP
  k_mlp<<<(BSZ * TT * NPT + 255) / 256, 256, 0, stream>>>(x, w1, w2, h);
  // 3) scatter-max
  k_scatter_bev<<<(BSZ * TT * NPT + 255) / 256, 256, 0, stream>>>(h, coord, (unsigned int*)bev);
  k_scatter_rv<<<(BSZ * NPT + 255) / 256, 256, 0, stream>>>(h, sph, (unsigned int*)rv);
  // 4) residual planes in place
  k_resid<<<4096, 256, 0, stream>>>(bev);
  // 5) WMMA convs: blocks = B * H * (W/64); 8 waves = 4 mtiles x 2 tile groups
  k_conv_wmma<192, BEVH, BEVW, true><<<BSZ * BEVH * (BEVW / 64), 256, 0, stream>>>(bev, pab, bevf);
  k_conv_wmma<64, RVH, RVW, false><<<BSZ * RVH * (RVW / 64), 256, 0, stream>>>(rv, par, rvf);
  // 6) gather + fuse + predict
  k_fuse<<<(BSZ * NPT + 255) / 256, 256, 0, stream>>>(h, bevf, rvf, coord, sph, wf, wp, out);
}